// Autoencoder_hourglass_channel_SPDnet_64630667870866
// MI455X (gfx1250) — compile-verified
//
#include <hip/hip_runtime.h>
#include <hip/hip_bf16.h>

// SPD autoencoder on MI455X (gfx1250, wave32).
// BiMap layers + Newton-Schulz-based ReEig, all mapped onto V_WMMA_F32_16X16X4_F32.
// BiMap LDS staging double-buffered via GLOBAL_LOAD_ASYNC_TO_LDS_B128 (ASYNCcnt).
// logeig->expeig->reeig sandwich collapses to identity (exp(log(x))=x, ReEig idempotent).

#define B_   2048
#define EPS_ 1e-4f

typedef float v2f __attribute__((ext_vector_type(2)));
typedef float v8f __attribute__((ext_vector_type(8)));

__device__ __forceinline__ v8f wmma_f32(v2f a, v2f b, v8f c) {
  // D(16x16,f32) += A(16x4,f32) x B(4x16,f32)
  return __builtin_amdgcn_wmma_f32_16x16x4_f32(false, a, false, b, (short)0, c,
                                               false, false);
}

// Async copy of 16 bytes global -> LDS per lane (tracked by ASYNCcnt).
__device__ __forceinline__ void async_b128(unsigned lds_off, const float* gptr) {
  asm volatile("global_load_async_to_lds_b128 %0, %1, off"
               :
               : "v"(lds_off), "v"((unsigned long long)(uintptr_t)gptr)
               : "memory");
}
__device__ __forceinline__ void wait_async0() {
  asm volatile("s_wait_asynccnt 0x0" ::: "memory");
}

// A-fragment (16x4) from row-major matrix S with leading dim ld at (row0, k0).
// ISA layout: lanes 0-15 hold M=0..15 / K={0,1}; lanes 16-31 hold K={2,3}.
__device__ __forceinline__ v2f fragA(const float* S, int ld, int row0, int k0, int lane) {
  int r = lane & 15, kk = (lane >> 4) << 1;
  v2f a;
  a.x = S[(row0 + r) * ld + k0 + kk];
  a.y = S[(row0 + r) * ld + k0 + kk + 1];
  return a;
}

// B-fragment (4x16) from row-major matrix S with leading dim ld at (k0, col0).
__device__ __forceinline__ v2f fragB(const float* S, int ld, int k0, int col0, int lane) {
  int c = lane & 15, kk = (lane >> 4) << 1;
  v2f b;
  b.x = S[(k0 + kk) * ld + col0 + c];
  b.y = S[(k0 + kk + 1) * ld + col0 + c];
  return b;
}

// A-fragment of S^T where S is row-major (K x Mrows) with leading dim ld.
__device__ __forceinline__ v2f fragAT(const float* S, int ld, int row0, int k0, int lane) {
  int r = lane & 15, kk = (lane >> 4) << 1;
  v2f a;
  a.x = S[(k0 + kk) * ld + row0 + r];
  a.y = S[(k0 + kk + 1) * ld + row0 + r];
  return a;
}

// Store 16x16 f32 accumulator: lane<16 -> N=lane, rows 0..7; lane>=16 -> rows 8..15.
__device__ __forceinline__ void storeC(float* D, int ld, int row0, int col0, v8f acc,
                                       int lane) {
  int c = lane & 15, mb = (lane >> 4) << 3;
#pragma unroll
  for (int r = 0; r < 8; ++r) D[(row0 + mb + r) * ld + col0 + c] = acc[r];
}

// C = A @ B, all NxN row-major in LDS, tiles distributed over 8 waves.
template <int N>
__device__ __forceinline__ void mm_nn(const float* A, const float* Bm, float* C,
                                      int wave, int lane) {
  constexpr int NT = N / 16;
  for (int t = wave; t < NT * NT; t += 8) {
    int ti = t / NT, tj = t % NT;
    v8f acc = {};
#pragma unroll
    for (int k0 = 0; k0 < N; k0 += 4)
      acc = wmma_f32(fragA(A, N, ti * 16, k0, lane), fragB(Bm, N, k0, tj * 16, lane), acc);
    storeC(C, N, ti * 16, tj * 16, acc, lane);
  }
}

// W = 1.5*Z - 0.5*(Z @ T)  (fused Newton-Schulz sign step)
template <int N>
__device__ __forceinline__ void mm_ns(const float* Z, const float* T, float* W,
                                      int wave, int lane) {
  constexpr int NT = N / 16;
  for (int t = wave; t < NT * NT; t += 8) {
    int ti = t / NT, tj = t % NT;
    v8f acc = {};
#pragma unroll
    for (int k0 = 0; k0 < N; k0 += 4)
      acc = wmma_f32(fragA(Z, N, ti * 16, k0, lane), fragB(T, N, k0, tj * 16, lane), acc);
    int c = lane & 15, mb = (lane >> 4) << 3;
#pragma unroll
    for (int r = 0; r < 8; ++r) {
      int row = ti * 16 + mb + r, col = tj * 16 + c;
      W[row * N + col] = 1.5f * Z[row * N + col] - 0.5f * acc[r];
    }
  }
}

// ---------------------------------------------------------------------------
// BiMap: Y[b,d] = sum_ci  W[d,ci]^T @ X[b,ci] @ W[d,ci]
// X: (B, CI, N, N)   W: (CO, CI, N, M)   Y: (B, CO, M, M)
// One workgroup (256 thr = 8 waves) per (b, d).
// Double-buffered async global->LDS staging overlaps channel ci+1's HBM
// traffic with channel ci's WMMA work.
// ---------------------------------------------------------------------------
template <int N, int M>
__global__ __launch_bounds__(256) void bimap_kernel(const float* __restrict__ X,
                                                    const float* __restrict__ Wm,
                                                    float* __restrict__ Y, int CI,
                                                    int CO) {
  __shared__ float Xs[2 * N * N];
  __shared__ float Wsh[2 * N * M];
  __shared__ float Ts[N * M];
  const int tid = threadIdx.x, wave = tid >> 5, lane = tid & 31;
  const int d = blockIdx.x % CO;
  const size_t b = blockIdx.x / CO;

  const unsigned xs_base = (unsigned)(uintptr_t)(void*)Xs;
  const unsigned ws_base = (unsigned)(uintptr_t)(void*)Wsh;

  constexpr int NT = N / 16, MT = M / 16;
  constexpr int OUT_TILES = MT * MT;
  constexpr int PER_WAVE = (OUT_TILES + 7) / 8;
  v8f zero = {};
  v8f acc[PER_WAVE];
#pragma unroll
  for (int p = 0; p < PER_WAVE; ++p) acc[p] = zero;

  // prologue: prefetch ci = 0 into buffer 0
  {
    const float* Xp = X + (b * CI) * (size_t)(N * N);
    const float* Wp = Wm + ((size_t)d * CI) * (size_t)(N * M);
    for (int i = tid; i < N * N / 4; i += 256) async_b128(xs_base + i * 16u, Xp + i * 4);
    for (int i = tid; i < N * M / 4; i += 256) async_b128(ws_base + i * 16u, Wp + i * 4);
  }

  int cur = 0;
  for (int ci = 0; ci < CI; ++ci) {
    wait_async0();       // this wave's copies into buf[cur] are in LDS
    __syncthreads();     // all waves' copies done; prior reads of buf[1-cur] done

    if (ci + 1 < CI) {   // prefetch next channel into the other buffer
      const float* Xp = X + (b * CI + ci + 1) * (size_t)(N * N);
      const float* Wp = Wm + ((size_t)d * CI + ci + 1) * (size_t)(N * M);
      unsigned xb = xs_base + (unsigned)(1 - cur) * (unsigned)(N * N * 4);
      unsigned wb = ws_base + (unsigned)(1 - cur) * (unsigned)(N * M * 4);
      for (int i = tid; i < N * N / 4; i += 256) async_b128(xb + i * 16u, Xp + i * 4);
      for (int i = tid; i < N * M / 4; i += 256) async_b128(wb + i * 16u, Wp + i * 4);
    }

    const float* Xc = Xs + cur * (N * N);
    const float* Wc = Wsh + cur * (N * M);

    // T = X @ W  (N x M)
    for (int t = wave; t < NT * MT; t += 8) {
      int ti = t / MT, tj = t % MT;
      v8f a = zero;
#pragma unroll
      for (int k0 = 0; k0 < N; k0 += 4)
        a = wmma_f32(fragA(Xc, N, ti * 16, k0, lane), fragB(Wc, M, k0, tj * 16, lane), a);
      storeC(Ts, M, ti * 16, tj * 16, a, lane);
    }
    __syncthreads();

    // out += W^T @ T  (M x M), K = N
    int p = 0;
    for (int t = wave; t < OUT_TILES; t += 8, ++p) {
      int ti = t / MT, tj = t % MT;
      v8f a = acc[p];
#pragma unroll
      for (int k0 = 0; k0 < N; k0 += 4)
        a = wmma_f32(fragAT(Wc, M, ti * 16, k0, lane), fragB(Ts, M, k0, tj * 16, lane), a);
      acc[p] = a;
    }
    cur ^= 1;
  }

  float* Yp = Y + (b * CO + d) * (size_t)(M * M);
  int p = 0;
  for (int t = wave; t < OUT_TILES; t += 8, ++p)
    storeC(Yp, M, (t / MT) * 16, (t % MT) * 16, acc[p], lane);
}

// ---------------------------------------------------------------------------
// ReEig(A) = 0.5*((A-eps I) + |A-eps I|) + eps I, in place on (grid.x, N, N).
// |M| = M*sign(M); sign via Newton-Schulz  Z <- 1.5 Z - 0.5 Z^3  after
// Frobenius normalization. All GEMMs on WMMA, matrices resident in LDS.
// ---------------------------------------------------------------------------
template <int N, int ITERS>
__global__ __launch_bounds__(256) void reeig_kernel(float* __restrict__ P) {
  __shared__ float Zs[N * N];
  __shared__ float Ts[N * N];
  __shared__ float Ws[N * N];
  __shared__ float s_scale;
  const int tid = threadIdx.x, wave = tid >> 5, lane = tid & 31;
  float* G = P + (size_t)blockIdx.x * (size_t)(N * N);

  // Z = A - eps I ; accumulate Frobenius norm^2
  float part = 0.f;
  for (int i = tid; i < N * N; i += 256) {
    float v = G[i];
    if ((i / N) == (i % N)) v -= EPS_;
    Zs[i] = v;
    part += v * v;
  }
  Ts[tid] = part;  // Ts doubles as reduction scratch before first GEMM
  __syncthreads();
  for (int off = 128; off > 0; off >>= 1) {
    if (tid < off) Ts[tid] += Ts[tid + off];
    __syncthreads();
  }
  if (tid == 0) s_scale = sqrtf(Ts[0]) + 1e-20f;
  __syncthreads();
  const float s = s_scale, inv = 1.f / s;
  for (int i = tid; i < N * N; i += 256) Zs[i] *= inv;  // Z = M0
  __syncthreads();

  // Newton-Schulz sign iteration
  for (int it = 0; it < ITERS; ++it) {
    mm_nn<N>(Zs, Zs, Ts, wave, lane);  // T = Z^2
    __syncthreads();
    mm_ns<N>(Zs, Ts, Ws, wave, lane);  // W = 1.5Z - 0.5 Z*T
    __syncthreads();
    for (int i = tid; i < N * N; i += 256) Zs[i] = Ws[i];
    __syncthreads();
  }

  // Reload M0 into Ws (Zs now holds S = sign(M0))
  for (int i = tid; i < N * N; i += 256) {
    float v = G[i];
    if ((i / N) == (i % N)) v -= EPS_;
    Ws[i] = v * inv;
  }
  __syncthreads();
  mm_nn<N>(Zs, Ws, Ts, wave, lane);  // T = S @ M0 ~= |M0|
  __syncthreads();

  // result = 0.5*s*(M0 + sym(T)) + eps I
  for (int i = tid; i < N * N; i += 256) {
    int r = i / N, c = i % N;
    float absv = 0.5f * (Ts[i] + Ts[c * N + r]);
    float v = 0.5f * s * (Ws[i] + absv);
    if (r == c) v += EPS_;
    G[i] = v;
  }
}

// ---------------------------------------------------------------------------
extern "C" void kernel_launch(void* const* d_in, const int* in_sizes, int n_in,
                              void* d_out, int out_size, void* d_ws, size_t ws_size,
                              hipStream_t stream) {
  const float* x  = (const float*)d_in[0];  // (B,4,64,64)
  const float* W1 = (const float*)d_in[1];  // (8,4,64,32)
  const float* W2 = (const float*)d_in[2];  // (4,8,32,16)
  const float* W3 = (const float*)d_in[3];  // (8,4,16,32)
  const float* W4 = (const float*)d_in[4];  // (4,8,32,64)
  float* out = (float*)d_out;               // (B,4,64,64)

  float* buf1 = (float*)d_ws;                          // (B,8,32,32) = 67 MB
  float* buf2 = buf1 + (size_t)B_ * 8 * 32 * 32;       // (B,4,16,16) = 8.4 MB

  // encoder
  bimap_kernel<64, 32><<<B_ * 8, 256, 0, stream>>>(x, W1, buf1, 4, 8);
  reeig_kernel<32, 12><<<B_ * 8, 256, 0, stream>>>(buf1);
  bimap_kernel<32, 16><<<B_ * 4, 256, 0, stream>>>(buf1, W2, buf2, 8, 4);
  reeig_kernel<16, 12><<<B_ * 4, 256, 0, stream>>>(buf2);
  // logeig -> expeig -> reeig on already-rectified code == identity (skipped)
  // decoder
  bimap_kernel<16, 32><<<B_ * 8, 256, 0, stream>>>(buf2, W3, buf1, 4, 8);
  reeig_kernel<32, 12><<<B_ * 8, 256, 0, stream>>>(buf1);
  bimap_kernel<32, 64><<<B_ * 4, 256, 0, stream>>>(buf1, W4, out, 8, 4);
  reeig_kernel<64, 12><<<B_ * 4, 256, 0, stream>>>(out);
}